// GraphConvBlock_57629871178304
// MI455X (gfx1250) — compile-verified
//
#include <hip/hip_runtime.h>
#include <math.h>

typedef float v2f __attribute__((ext_vector_type(2)));
typedef float v8f __attribute__((ext_vector_type(8)));

#define EPSF 1e-5f

// ---------------------------------------------------------------------------
// Kernel 1: nodes = relu( BN( Wn @ x + bn_b ) )     per batch: 64x64 @ 64x1024
// One wave per 16x16 output tile, K=64 via 16 x V_WMMA_F32_16X16X4_F32.
// ---------------------------------------------------------------------------
__global__ void k1_nodes(const float* __restrict__ x, const float* __restrict__ Wn,
                         const float* __restrict__ bn_b, const float* __restrict__ gn,
                         const float* __restrict__ betan, const float* __restrict__ mn,
                         const float* __restrict__ vn, float* __restrict__ nodes) {
  int wid  = blockIdx.x * 8 + (threadIdx.x >> 5);
  int lane = threadIdx.x & 31;
  int b    = wid >> 8;                 // 256 tiles per batch
  int rem  = wid & 255;
  int m0   = (rem >> 6) << 4;          // 4 row tiles
  int n0   = (rem & 63) << 4;          // 64 col tiles
  int half = lane >> 4;
  int l    = lane & 15;

  const float* xb = x + b * 64 * 1024;
  v8f acc = {};
  for (int k0 = 0; k0 < 64; k0 += 4) {
    int ka = k0 + 2 * half;            // A lanes<16: K=0,1 ; lanes>=16: K=2,3
    v2f a, bb;
    a.x  = Wn[(m0 + l) * 64 + ka];
    a.y  = Wn[(m0 + l) * 64 + ka + 1];
    bb.x = xb[ka * 1024 + n0 + l];     // B row K striped across lanes
    bb.y = xb[(ka + 1) * 1024 + n0 + l];
    acc = __builtin_amdgcn_wmma_f32_16x16x4_f32(false, a, false, bb,
                                                (short)0, acc, false, false);
  }
  float* nb = nodes + b * 64 * 1024;
  #pragma unroll
  for (int r = 0; r < 8; ++r) {        // C/D: vgpr r -> row r (lanes<16) / r+8
    int m = m0 + r + 8 * half;
    float s = gn[m] * rsqrtf(vn[m] + EPSF);
    float v = (acc[r] + bn_b[m] - mn[m]) * s + betan[m];
    nb[m * 1024 + n0 + l] = fmaxf(v, 0.f);
  }
}

// ---------------------------------------------------------------------------
// Kernel 2: S = [Wa;Wb] @ nodes (64x64 @ 64x1024), epilogue folds edge-BN:
//   U1[n][m] = s1*sc + ((be1-me)*sc + bete)   (rows 0..31)
//   U2[n][m] = s2*sc                          (rows 32..63)
// ---------------------------------------------------------------------------
__global__ void k2_edges(const float* __restrict__ nodes, const float* __restrict__ We1,
                         const float* __restrict__ be1, const float* __restrict__ ge,
                         const float* __restrict__ bete, const float* __restrict__ me,
                         const float* __restrict__ ve,
                         float* __restrict__ U1, float* __restrict__ U2) {
  int wid  = blockIdx.x * 8 + (threadIdx.x >> 5);
  int lane = threadIdx.x & 31;
  int b    = wid >> 8;
  int rem  = wid & 255;
  int m0   = (rem >> 6) << 4;
  int n0   = (rem & 63) << 4;
  int half = lane >> 4;
  int l    = lane & 15;

  const float* nb = nodes + b * 64 * 1024;
  int row = m0 + l;                    // stacked matrix: row<32 -> Wa, else Wb
  const float* wrow = (row < 32) ? (We1 + row * 128)
                                 : (We1 + (row - 32) * 128 + 64);
  v8f acc = {};
  for (int k0 = 0; k0 < 64; k0 += 4) {
    int ka = k0 + 2 * half;
    v2f a, bb;
    a.x  = wrow[ka];
    a.y  = wrow[ka + 1];
    bb.x = nb[ka * 1024 + n0 + l];
    bb.y = nb[(ka + 1) * 1024 + n0 + l];
    acc = __builtin_amdgcn_wmma_f32_16x16x4_f32(false, a, false, bb,
                                                (short)0, acc, false, false);
  }
  int n = n0 + l;
  #pragma unroll
  for (int r = 0; r < 8; ++r) {
    int m = m0 + r + 8 * half;         // tile-uniform branch (m0 multiple of 16)
    if (m < 32) {
      float sc  = ge[m] * rsqrtf(ve[m] + EPSF);
      float cst = (be1[m] - me[m]) * sc + bete[m];
      U1[(b * 1024 + n) * 32 + m] = acc[r] * sc + cst;
    } else {
      int mm = m - 32;
      float sc = ge[mm] * rsqrtf(ve[mm] + EPSF);
      U2[(b * 1024 + n) * 32 + mm] = acc[r] * sc;
    }
  }
}

// ---------------------------------------------------------------------------
// Kernel 3: E[i][j] = exp(sigmoid( sum_m relu(u1[i,m]+u2[j,m]) * We2[m] + be2 ))
// ReLU inside the contraction -> not a GEMM; tiled VALU kernel, LDS-staged.
// Block tile: 16 i-rows x 64 j-cols, 4 pairs per thread.
// ---------------------------------------------------------------------------
__global__ void k3_pair(const float* __restrict__ U1, const float* __restrict__ U2,
                        const float* __restrict__ We2, const float* __restrict__ be2,
                        float* __restrict__ E) {
  __shared__ float sU1[16 * 33];       // +1 pad -> conflict-free (gcd(33,64)=1)
  __shared__ float sU2[64 * 33];
  __shared__ float sW[32];
  int t   = threadIdx.x;
  int bid = blockIdx.x;
  int b   = bid >> 10;                 // 1024 blocks per batch
  int rem = bid & 1023;
  int i0  = (rem >> 4) << 4;           // 64 i-tiles of 16
  int j0  = (rem & 15) << 6;           // 16 j-tiles of 64

  for (int idx = t; idx < 16 * 32; idx += 256) {
    int i = idx >> 5, m = idx & 31;
    sU1[i * 33 + m] = U1[(b * 1024 + i0 + i) * 32 + m];
  }
  for (int idx = t; idx < 64 * 32; idx += 256) {
    int j = idx >> 5, m = idx & 31;
    sU2[j * 33 + m] = U2[(b * 1024 + j0 + j) * 32 + m];
  }
  if (t < 32) sW[t] = We2[t];
  __syncthreads();

  float b2 = be2[0];
  #pragma unroll
  for (int p = 0; p < 4; ++p) {
    int lin = t + 256 * p;
    int il = lin >> 6, jl = lin & 63;
    const float* u1 = sU1 + il * 33;
    const float* u2 = sU2 + jl * 33;
    float acc = 0.f;
    #pragma unroll
    for (int m = 0; m < 32; ++m)
      acc += fmaxf(u1[m] + u2[m], 0.f) * sW[m];
    float logit = acc + b2;
    float sig = 1.f / (1.f + expf(-logit));     // sigmoid in (0,1): softmax
    E[(size_t)(b * 1024 + i0 + il) * 1024 + j0 + jl] = expf(sig); // needs no max-shift
  }
}

// ---------------------------------------------------------------------------
// Kernel 3b: RINV[row] = 1 / sum_j E[row][j]   (wave32 shuffle reduction)
// ---------------------------------------------------------------------------
__global__ void k3b_rowsum(const float* __restrict__ E, float* __restrict__ RINV) {
  int row  = blockIdx.x * 8 + (threadIdx.x >> 5);   // 2048 rows
  int lane = threadIdx.x & 31;
  const float* er = E + (size_t)row * 1024;
  float s = 0.f;
  for (int k = lane; k < 1024; k += 32) s += er[k];
  #pragma unroll
  for (int off = 16; off > 0; off >>= 1)
    s += __shfl_xor(s, off, 32);
  if (lane == 0) RINV[row] = 1.f / s;
}

// ---------------------------------------------------------------------------
// Kernel 4: out = nodes @ (E * rinv)   per batch: 64x1024 @ 1024x1024
// Softmax normalization folded into B-operand load; 256 WMMA k-steps/tile.
// ---------------------------------------------------------------------------
__global__ void k4_out(const float* __restrict__ nodes, const float* __restrict__ E,
                       const float* __restrict__ RINV, float* __restrict__ out) {
  int wid  = blockIdx.x * 8 + (threadIdx.x >> 5);
  int lane = threadIdx.x & 31;
  int b    = wid >> 8;
  int rem  = wid & 255;
  int m0   = (rem >> 6) << 4;
  int n0   = (rem & 63) << 4;
  int half = lane >> 4;
  int l    = lane & 15;

  const float* nb = nodes + b * 65536;
  const float* eb = E + (size_t)b * 1024 * 1024;
  const float* rb = RINV + b * 1024;
  v8f acc = {};
  for (int k0 = 0; k0 < 1024; k0 += 4) {
    int ka = k0 + 2 * half;
    v2f a, bb;
    a.x  = nb[(m0 + l) * 1024 + ka];
    a.y  = nb[(m0 + l) * 1024 + ka + 1];
    bb.x = eb[(size_t)ka * 1024 + n0 + l] * rb[ka];
    bb.y = eb[(size_t)(ka + 1) * 1024 + n0 + l] * rb[ka + 1];
    acc = __builtin_amdgcn_wmma_f32_16x16x4_f32(false, a, false, bb,
                                                (short)0, acc, false, false);
  }
  float* ob = out + b * 65536;
  #pragma unroll
  for (int r = 0; r < 8; ++r) {
    int m = m0 + r + 8 * half;
    ob[m * 1024 + n0 + l] = acc[r];
  }
}

// ---------------------------------------------------------------------------
extern "C" void kernel_launch(void* const* d_in, const int* in_sizes, int n_in,
                              void* d_out, int out_size, void* d_ws, size_t ws_size,
                              hipStream_t stream) {
  const float* x     = (const float*)d_in[0];
  const float* Wn    = (const float*)d_in[1];
  const float* bn_b  = (const float*)d_in[2];
  const float* gn    = (const float*)d_in[3];
  const float* betan = (const float*)d_in[4];
  const float* mn    = (const float*)d_in[5];
  const float* vn    = (const float*)d_in[6];
  const float* We1   = (const float*)d_in[7];
  const float* be1   = (const float*)d_in[8];
  const float* ge    = (const float*)d_in[9];
  const float* bete  = (const float*)d_in[10];
  const float* me    = (const float*)d_in[11];
  const float* ve    = (const float*)d_in[12];
  const float* We2   = (const float*)d_in[13];
  const float* be2   = (const float*)d_in[14];
  float* out = (float*)d_out;

  // workspace layout (floats)
  float* ws    = (float*)d_ws;
  float* NODES = ws;                 // [2][64][1024]   131072
  float* U1    = ws + 131072;        // [2][1024][32]    65536
  float* U2    = ws + 196608;        // [2][1024][32]    65536
  float* E     = ws + 262144;        // [2][1024][1024] 2097152
  float* RINV  = ws + 2359296;       // [2][1024]         2048
  (void)in_sizes; (void)n_in; (void)out_size; (void)ws_size;

  k1_nodes  <<<64,   256, 0, stream>>>(x, Wn, bn_b, gn, betan, mn, vn, NODES);
  k2_edges  <<<64,   256, 0, stream>>>(NODES, We1, be1, ge, bete, me, ve, U1, U2);
  k3_pair   <<<2048, 256, 0, stream>>>(U1, U2, We2, be2, E);
  k3b_rowsum<<<256,  256, 0, stream>>>(E, RINV);
  k4_out    <<<64,   256, 0, stream>>>(NODES, E, RINV, out);
}